// GCNNet_17918603559053
// MI455X (gfx1250) — compile-verified
//
#include <hip/hip_runtime.h>

typedef __attribute__((ext_vector_type(2))) float v2f;
typedef __attribute__((ext_vector_type(8))) float v8f;

// ---------------------------------------------------------------- utilities
__global__ void zero_kernel(float* __restrict__ p, size_t n) {
    size_t i = (size_t)blockIdx.x * blockDim.x + threadIdx.x;
    size_t stride = (size_t)gridDim.x * blockDim.x;
    for (; i < n; i += stride) p[i] = 0.0f;
}

__global__ void degree_kernel(const int* __restrict__ src, const int* __restrict__ dst,
                              float* __restrict__ deg_out, float* __restrict__ deg_in,
                              int n_edges) {
    int e = blockIdx.x * blockDim.x + threadIdx.x;
    if (e < n_edges) {
        atomicAdd(&deg_out[src[e]], 1.0f);
        atomicAdd(&deg_in[dst[e]], 1.0f);
    }
}

__global__ void normify_kernel(float* __restrict__ p, int n) {
    int i = blockIdx.x * blockDim.x + threadIdx.x;
    if (i < n) {
        float d = p[i];
        p[i] = rsqrtf(fmaxf(d, 1.0f));
    }
}

// ------------------------------------------------- WMMA GEMM  Y = (X*s) @ W
// X: [n_rows,128] fp32 (row-major), W: [128,NDIM] fp32, Y: [n_rows,NDIM]
// Optional per-row scale (rowscale != nullptr). One wave -> 16 output rows.
template <int NDIM>
__global__ __launch_bounds__(256) void gemm_wmma_f32(
    const float* __restrict__ X, const float* __restrict__ rowscale,
    const float* __restrict__ W, float* __restrict__ Y, int n_rows) {
    constexpr int K  = 128;
    constexpr int NT = NDIM / 16;

    const int wave = threadIdx.x >> 5;               // 0..7
    const int lane = threadIdx.x & 31;
    const int row0 = (blockIdx.x * 8 + wave) * 16;   // wave-uniform
    if (row0 >= n_rows) return;                      // uniform exit, EXEC stays full

    const int m     = lane & 15;
    const int khalf = (lane >> 4) * 2;               // 0 or 2 (K split by lane half)
    int r  = row0 + m;
    int rc = (r < n_rows) ? r : (n_rows - 1);        // clamp A reads for tail rows
    const float scale = rowscale ? rowscale[rc] : 1.0f;
    const float* __restrict__ xrow = X + (size_t)rc * K;

    v8f acc[NT];
#pragma unroll
    for (int t = 0; t < NT; ++t) acc[t] = (v8f)0.0f;

    for (int k0 = 0; k0 < K; k0 += 4) {
        v2f a;
        a.x = xrow[k0 + khalf + 0] * scale;
        a.y = xrow[k0 + khalf + 1] * scale;
#pragma unroll
        for (int t = 0; t < NT; ++t) {
            const int n = t * 16 + m;
            v2f b;
            b.x = W[(size_t)(k0 + khalf + 0) * NDIM + n];
            b.y = W[(size_t)(k0 + khalf + 1) * NDIM + n];
            acc[t] = __builtin_amdgcn_wmma_f32_16x16x4_f32(
                false, a, false, b, (short)0, acc[t], false, false);
        }
    }

    // D layout: VGPR g on lane l -> row (g + 8*(l>>4)), col (l&15)
    const int rowhi = (lane >> 4) * 8;
#pragma unroll
    for (int t = 0; t < NT; ++t) {
#pragma unroll
        for (int g = 0; g < 8; ++g) {
            int rr = row0 + g + rowhi;
            if (rr < n_rows) Y[(size_t)rr * NDIM + t * 16 + m] = acc[t][g];
        }
    }
}

// --------------------------------------------- edge scatter:  agg[dst] += m[src]
// DIM/4 lanes per edge, float4 gather + 4 scalar f32 atomics per lane.
template <int DIM>
__global__ __launch_bounds__(256) void scatter_add_kernel(
    const float* __restrict__ msg, const int* __restrict__ src,
    const int* __restrict__ dst, float* __restrict__ agg, int n_edges) {
    constexpr int CHUNKS = DIM / 4;
    unsigned long long gid = (unsigned long long)blockIdx.x * blockDim.x + threadIdx.x;
    unsigned long long e = gid / CHUNKS;
    int c = (int)(gid % CHUNKS);
    if (e >= (unsigned long long)n_edges) return;
    int s = src[e], d = dst[e];
    const float4 v = ((const float4*)(msg + (size_t)s * DIM))[c];
    float* out = agg + (size_t)d * DIM + c * 4;
    atomicAdd(out + 0, v.x);
    atomicAdd(out + 1, v.y);
    atomicAdd(out + 2, v.z);
    atomicAdd(out + 3, v.w);
}

// ------------------------------------------------- epilogues
// h1s = relu(agg*norm_d + b) * norm_s   (prepares scaled layer-2 input)
__global__ void finalize_relu_scale_kernel(const float* __restrict__ agg,
                                           const float* __restrict__ norm_d,
                                           const float* __restrict__ norm_s,
                                           const float* __restrict__ bias,
                                           float* __restrict__ out, int n_nodes) {
    constexpr int DIM = 128;
    size_t idx = (size_t)blockIdx.x * blockDim.x + threadIdx.x;
    if (idx >= (size_t)n_nodes * DIM) return;
    int i = (int)(idx / DIM), j = (int)(idx % DIM);
    float v = agg[idx] * norm_d[i] + bias[j];
    v = fmaxf(v, 0.0f);
    out[idx] = v * norm_s[i];
}

// out = out*norm_d + b   (in place on d_out)
__global__ void finalize_out_kernel(float* __restrict__ out,
                                    const float* __restrict__ norm_d,
                                    const float* __restrict__ bias, int n_nodes) {
    constexpr int DIM = 64;
    size_t idx = (size_t)blockIdx.x * blockDim.x + threadIdx.x;
    if (idx >= (size_t)n_nodes * DIM) return;
    int i = (int)(idx / DIM), j = (int)(idx % DIM);
    out[idx] = out[idx] * norm_d[i] + bias[j];
}

// ---------------------------------------------------------------- launcher
extern "C" void kernel_launch(void* const* d_in, const int* in_sizes, int n_in,
                              void* d_out, int out_size, void* d_ws, size_t ws_size,
                              hipStream_t stream) {
    const float* features = (const float*)d_in[0];
    const int*   src      = (const int*)d_in[1];
    const int*   dst      = (const int*)d_in[2];
    const float* W1       = (const float*)d_in[3];
    const float* b1       = (const float*)d_in[4];
    const float* W2       = (const float*)d_in[5];
    const float* b2       = (const float*)d_in[6];
    float*       out      = (float*)d_out;

    const int IN_DIM = 128, HID = 128, OUT = 64;
    const int n_nodes = in_sizes[0] / IN_DIM;
    const int n_edges = in_sizes[1];

    // workspace layout: [norm_s | norm_d | bufA (n*128) | bufB (n*128)]
    char* ws = (char*)d_ws;
    float* norm_s = (float*)ws;                 // n_nodes
    float* norm_d = norm_s + n_nodes;           // n_nodes
    size_t ofsA = (((size_t)2 * n_nodes * sizeof(float)) + 255) & ~(size_t)255;
    float* bufA = (float*)(ws + ofsA);                                        // n*128
    float* bufB = (float*)(ws + ofsA + (size_t)n_nodes * HID * sizeof(float)); // n*128

    const int TPB = 256;

    // 1) degrees -> norms (norm_s/norm_d are contiguous: one pass each)
    zero_kernel<<<512, TPB, 0, stream>>>(norm_s, (size_t)2 * n_nodes);
    degree_kernel<<<(n_edges + TPB - 1) / TPB, TPB, 0, stream>>>(src, dst, norm_s, norm_d, n_edges);
    normify_kernel<<<(2 * n_nodes + TPB - 1) / TPB, TPB, 0, stream>>>(norm_s, 2 * n_nodes);

    // 2) layer 1 GEMM: m1 = (features * norm_s) @ W1  -> bufA
    int rowblocks = (n_nodes + 15) / 16;
    int gblocks   = (rowblocks + 7) / 8;
    gemm_wmma_f32<128><<<gblocks, TPB, 0, stream>>>(features, norm_s, W1, bufA, n_nodes);

    // 3) scatter-add over edges -> bufB (zeroed)
    zero_kernel<<<4096, TPB, 0, stream>>>(bufB, (size_t)n_nodes * HID);
    {
        unsigned long long total = (unsigned long long)n_edges * (HID / 4);
        unsigned int blocks = (unsigned int)((total + TPB - 1) / TPB);
        scatter_add_kernel<128><<<blocks, TPB, 0, stream>>>(bufA, src, dst, bufB, n_edges);
    }

    // 4) h1s = relu(agg*norm_d + b1) * norm_s  -> bufA (m1 dead)
    {
        size_t total = (size_t)n_nodes * HID;
        finalize_relu_scale_kernel<<<(unsigned int)((total + TPB - 1) / TPB), TPB, 0, stream>>>(
            bufB, norm_d, norm_s, b1, bufA, n_nodes);
    }

    // 5) layer 2 GEMM: m2 = h1s @ W2 -> bufB (agg1 dead)
    gemm_wmma_f32<64><<<gblocks, TPB, 0, stream>>>(bufA, nullptr, W2, bufB, n_nodes);

    // 6) scatter-add layer 2 directly into d_out (zeroed)
    zero_kernel<<<4096, TPB, 0, stream>>>(out, (size_t)n_nodes * OUT);
    {
        unsigned long long total = (unsigned long long)n_edges * (OUT / 4);
        unsigned int blocks = (unsigned int)((total + TPB - 1) / TPB);
        scatter_add_kernel<64><<<blocks, TPB, 0, stream>>>(bufB, src, dst, out, n_edges);
    }

    // 7) out = out*norm_d + b2 (in place)
    {
        size_t total = (size_t)n_nodes * OUT;
        finalize_out_kernel<<<(unsigned int)((total + TPB - 1) / TPB), TPB, 0, stream>>>(
            out, norm_d, b2, n_nodes);
    }
}